// ConvolutionLayer_4784593568029
// MI455X (gfx1250) — compile-verified
//
#include <hip/hip_runtime.h>
#include <hip/hip_bf16.h>

typedef __attribute__((ext_vector_type(2))) float v2f;
typedef __attribute__((ext_vector_type(8))) float v8f;

// ---------------------------------------------------------------------------
// 1) zero u32 array
// ---------------------------------------------------------------------------
__global__ void zero_u32(unsigned* __restrict__ p, int n) {
    int i = (int)blockIdx.x * 256 + (int)threadIdx.x;
    if (i < n) p[i] = 0u;
}

// ---------------------------------------------------------------------------
// 2) per-destination degree histogram (int atomics only)
// ---------------------------------------------------------------------------
__global__ void edge_degree(const long long* __restrict__ ei,
                            unsigned* __restrict__ deg, int E) {
    int i = (int)blockIdx.x * 256 + (int)threadIdx.x;
    if (i >= E) return;
    int dst = (int)ei[(size_t)E + i];
    atomicAdd(&deg[dst], 1u);
}

// ---------------------------------------------------------------------------
// 3) exclusive prefix scan of degrees -> CSR offsets (+ cursor copy)
// ---------------------------------------------------------------------------
__global__ void scan_offsets(const unsigned* __restrict__ deg,
                             unsigned* __restrict__ offs,
                             unsigned* __restrict__ cursor, int N) {
    __shared__ unsigned sh[1024];
    __shared__ unsigned carry;
    const int t = (int)threadIdx.x;
    if (t == 0) carry = 0u;
    __syncthreads();
    for (int base = 0; base < N; base += 1024) {
        int i = base + t;
        unsigned v = (i < N) ? deg[i] : 0u;
        sh[t] = v;
        __syncthreads();
        #pragma unroll
        for (int off = 1; off < 1024; off <<= 1) {
            unsigned add = (t >= off) ? sh[t - off] : 0u;
            __syncthreads();
            sh[t] += add;
            __syncthreads();
        }
        unsigned incl = sh[t];
        unsigned excl = incl - v;
        if (i < N) {
            unsigned o = carry + excl;
            offs[i] = o;
            cursor[i] = o;
        }
        __syncthreads();
        if (t == 1023) carry += incl;
        __syncthreads();
    }
    if (t == 0) offs[N] = carry;
}

// ---------------------------------------------------------------------------
// 4) CSR build: place each edge into its destination bucket
// ---------------------------------------------------------------------------
__global__ void build_csr(const long long* __restrict__ ei,
                          const float* __restrict__ ev,
                          unsigned* __restrict__ cursor,
                          int* __restrict__ ssrc,
                          float* __restrict__ sval, int E) {
    int i = (int)blockIdx.x * 256 + (int)threadIdx.x;
    if (i >= E) return;
    int src = (int)ei[i];
    int dst = (int)ei[(size_t)E + i];
    unsigned p = atomicAdd(&cursor[dst], 1u);
    ssrc[p] = src;
    sval[p] = ev[i];
}

// ---------------------------------------------------------------------------
// 5) dual fp32 WMMA GEMM:
//    phase 0: out = X @ W0 + bias      phase 1: s1 = X @ W1
//    Weights staged in LDS pre-packed into B-fragment order: pair-row p holds
//    {W[2p][c], W[2p+1][c]} as one float2, so every WMMA B operand is a single
//    ds_load_b64 into an even-aligned VGPR pair (no assembly movs).
//    Pair-row stride 144 -> half0 lanes hit dword banks b..b+31, half1 lanes
//    b+32..b+63: conflict-free.
//    n-tiles done in 2 groups of 4 (runtime loops) to keep VGPRs ~128.
// ---------------------------------------------------------------------------
__global__ __launch_bounds__(256) void gemm_dual(
    const float* __restrict__ X, const float* __restrict__ W0,
    const float* __restrict__ W1, const float* __restrict__ bias,
    float* __restrict__ out, float* __restrict__ s1, int N)
{
    constexpr int PSTR = 144;                 // pair-row stride (float2 units)
    __shared__ v2f bshr[64 * PSTR];           // 73,728 B  (<< 320 KB WGP LDS)

    const int t    = (int)threadIdx.x;
    const int lane = t & 31;
    const int wave = t >> 5;
    const int half = lane >> 4;               // 0: lanes 0-15, 1: lanes 16-31
    const int lid  = lane & 15;
    const int tile = (int)blockIdx.x * 8 + wave;
    const bool active = tile < (N >> 4);      // wave-uniform -> EXEC all-ones
    const int row0 = tile << 4;

    // Preload this wave's A strip: 32 k-steps of 16x4, v2f per lane.
    // A layout: lanes 0-15 hold K={4k,4k+1}, lanes 16-31 hold K={4k+2,4k+3}.
    v2f afrag[32];
    if (active) {
        const float* ap = X + (size_t)(row0 + lid) * 128 + 2 * half;
        #pragma unroll
        for (int ks = 0; ks < 32; ++ks)
            afrag[ks] = *(const v2f*)(ap + 4 * ks);
    }

    #pragma unroll 1
    for (int phase = 0; phase < 2; ++phase) {
        const float* W = phase ? W1 : W0;

        // stage weights pre-packed: bshr[p*PSTR + c] = {W[2p][c], W[2p+1][c]}
        for (int i = t; i < 64 * 128; i += 256) {
            int p = i >> 7;                   // pair-row 0..63
            int c = i & 127;                  // column 0..127
            v2f w;
            w.x = W[(size_t)(2 * p) * 128 + c];
            w.y = W[(size_t)(2 * p + 1) * 128 + c];
            bshr[p * PSTR + c] = w;
        }
        __syncthreads();

        if (active) {
            #pragma unroll 1
            for (int g = 0; g < 2; ++g) {     // two groups of 4 n-tiles
                v8f z = {};
                v8f acc[4];
                #pragma unroll
                for (int nt = 0; nt < 4; ++nt) acc[nt] = z;

                const v2f* bb = &bshr[half * PSTR + g * 64 + lid];
                #pragma unroll
                for (int ks = 0; ks < 32; ++ks) {
                    const v2f* bk = bb + (size_t)(2 * ks) * PSTR;
                    #pragma unroll
                    for (int nt = 0; nt < 4; ++nt) {   // 4 independent chains
                        v2f b = bk[nt * 16];           // one ds_load_b64
                        acc[nt] = __builtin_amdgcn_wmma_f32_16x16x4_f32(
                            false, afrag[ks], false, b, (short)0, acc[nt],
                            false, false);
                    }
                }

                // C/D layout: VGPR r -> row r (lanes 0-15) / r+8 (lanes 16-31)
                if (phase == 0) {
                    #pragma unroll
                    for (int nt = 0; nt < 4; ++nt) {
                        const int col = (g * 4 + nt) * 16 + lid;
                        const float bv = bias[col];
                        #pragma unroll
                        for (int r = 0; r < 8; ++r)
                            out[(size_t)(row0 + r + 8 * half) * 128 + col] =
                                acc[nt][r] + bv;
                    }
                } else {
                    #pragma unroll
                    for (int nt = 0; nt < 4; ++nt) {
                        const int col = (g * 4 + nt) * 16 + lid;
                        #pragma unroll
                        for (int r = 0; r < 8; ++r)
                            s1[(size_t)(row0 + r + 8 * half) * 128 + col] =
                                acc[nt][r];
                    }
                }
            }
        }
        __syncthreads();
    }
}

// ---------------------------------------------------------------------------
// 6) CSR gather: one wave per destination node, float4 per lane (32*4 = 128),
//    register accumulation, single non-atomic RMW of out.
//    support1 (51 MB) is L2-resident on the 192 MB L2.
// ---------------------------------------------------------------------------
__global__ __launch_bounds__(256) void gather_agg(
    const float* __restrict__ s1, const unsigned* __restrict__ offs,
    const int* __restrict__ ssrc, const float* __restrict__ sval,
    float* __restrict__ out, int N)
{
    const int node = (int)blockIdx.x * 8 + ((int)threadIdx.x >> 5);
    if (node >= N) return;
    const int lane = (int)threadIdx.x & 31;

    const unsigned beg = offs[node];
    const unsigned end = offs[node + 1];
    const float4* s14 = (const float4*)s1;

    float4 acc = make_float4(0.f, 0.f, 0.f, 0.f);
    for (unsigned j = beg; j < end; ++j) {
        const int src = ssrc[j];
        const float v = sval[j];
        if (j + 1 < end)  // prefetch next source row (global_prefetch_b8)
            __builtin_prefetch(&s14[(size_t)ssrc[j + 1] * 32 + lane], 0, 0);
        const float4 x = s14[(size_t)src * 32 + lane];
        acc.x += v * x.x; acc.y += v * x.y;
        acc.z += v * x.z; acc.w += v * x.w;
    }

    float4* o4 = (float4*)out;
    float4 c = o4[(size_t)node * 32 + lane];
    c.x += acc.x; c.y += acc.y; c.z += acc.z; c.w += acc.w;
    o4[(size_t)node * 32 + lane] = c;
}

// ---------------------------------------------------------------------------
// launch
// ---------------------------------------------------------------------------
extern "C" void kernel_launch(void* const* d_in, const int* in_sizes, int n_in,
                              void* d_out, int out_size, void* d_ws, size_t ws_size,
                              hipStream_t stream)
{
    const float*     X    = (const float*)d_in[0];
    const long long* ei   = (const long long*)d_in[1];
    const float*     ev   = (const float*)d_in[2];
    const float*     W0   = (const float*)d_in[3];
    const float*     W1   = (const float*)d_in[4];
    const float*     bias = (const float*)d_in[5];
    float* out = (float*)d_out;

    const int D  = 128;
    const int ND = in_sizes[0];      // N*D
    const int N  = ND / D;
    const int E  = in_sizes[2];

    // workspace layout (~65 MB): s1 | deg | offs | cursor | ssrc | sval
    char* ws = (char*)d_ws;
    float*    s1     = (float*)ws;                         // N*D floats
    unsigned* deg    = (unsigned*)(ws + (size_t)ND * 4);   // N
    unsigned* offs   = deg + N;                            // N+1
    unsigned* cursor = offs + (N + 1);                     // N
    int*      ssrc   = (int*)(cursor + N);                 // E
    float*    sval   = (float*)(ssrc + E);                 // E

    zero_u32    <<<(N + 255) / 256, 256, 0, stream>>>(deg, N);
    edge_degree <<<(E + 255) / 256, 256, 0, stream>>>(ei, deg, E);
    scan_offsets<<<1, 1024, 0, stream>>>(deg, offs, cursor, N);
    build_csr   <<<(E + 255) / 256, 256, 0, stream>>>(ei, ev, cursor, ssrc, sval, E);

    const int tiles = N / 16;                 // N = 100000 -> 6250 strips
    gemm_dual   <<<(tiles + 7) / 8, 256, 0, stream>>>(X, W0, W1, bias, out, s1, N);
    gather_agg  <<<(N + 7) / 8,     256, 0, stream>>>(s1, offs, ssrc, sval, out, N);
}